// KnowledgeDistillationKLFeatLevelLoss_74182675136878
// MI455X (gfx1250) — compile-verified
//
#include <hip/hip_runtime.h>

// ---------------------------------------------------------------------------
// Problem constants (from reference): shape (32, 256, 64, 64) fp32
// ---------------------------------------------------------------------------
#define NB 32                      // batch
#define CHW (256 * 64 * 64)        // 1048576 elements per sample
#define BPS 64                     // blocks per sample
#define TPB 256                    // threads per block (8 waves of 32)
#define V4_PER_SAMPLE (CHW / 4)    // 262144 float4 per sample
#define V4_PER_BLOCK (V4_PER_SAMPLE / BPS) // 4096 float4 per block
#define ITERS (V4_PER_BLOCK / TPB) // 16 float4 per thread
#define EPS 1e-5f
#define BETA 2.0f

// ws layout (floats): [0,32) sum | [32,64) sumsq | [64,96) mean
//                     [96,128) rstd | [128,160) per-sample loss
#define WS_SUM   0
#define WS_SQ    32
#define WS_MEAN  64
#define WS_RSTD  96
#define WS_LOSS  128
#define WS_FLOATS 160

typedef __attribute__((ext_vector_type(2))) float v2f;
typedef __attribute__((ext_vector_type(8))) float v8f;

// ---------------------------------------------------------------------------
// Exact-fp32 full-wave (32 lane) reduction using V_WMMA_F32_16X16X4_F32.
// A: a0 = partial, a1 = 0  ->  row m of A = {p[m], 0, p[m+16], 0}
// B: all ones (4x16)       ->  D[m][n] = p[m] + p[m+16]  (same for all n)
// D layout: lane L, vgpr v holds D[v + 8*(L>=16)][L%16]; summing the 8 vgprs
// gives half the rows, shfl_xor(16) adds the other half -> every lane gets
// the full 32-lane sum. Requires EXEC == all ones (full wave, no divergence).
// ---------------------------------------------------------------------------
__device__ __forceinline__ float wave_reduce_wmma(float p) {
  v2f a; a[0] = p;    a[1] = 0.0f;
  v2f b; b[0] = 1.0f; b[1] = 1.0f;
  v8f c = {};
  c = __builtin_amdgcn_wmma_f32_16x16x4_f32(
      /*neg_a=*/false, a, /*neg_b=*/false, b,
      /*c_mod=*/(short)0, c, /*reuse_a=*/false, /*reuse_b=*/false);
  float s = ((c[0] + c[1]) + (c[2] + c[3])) + ((c[4] + c[5]) + (c[6] + c[7]));
  s += __shfl_xor(s, 16, 32);
  return s;
}

// ---------------------------------------------------------------------------
__global__ void k_init_ws(float* __restrict__ ws) {
  int i = threadIdx.x;
  if (i < WS_FLOATS) ws[i] = 0.0f;
}

// Pass 1: per-sample sum and sum-of-squares of teacher (streaming b128 loads)
__global__ void k_stats(const float4* __restrict__ t, float* __restrict__ ws) {
  const int s = blockIdx.x / BPS;
  const int j = blockIdx.x % BPS;
  const int base = s * V4_PER_SAMPLE + j * V4_PER_BLOCK + (int)threadIdx.x;

  float sum = 0.0f, sq = 0.0f;
#pragma unroll
  for (int k = 0; k < ITERS; ++k) {
    const float4 v = t[base + k * TPB];
    sum += (v.x + v.y) + (v.z + v.w);
    sq  += (v.x * v.x + v.y * v.y) + (v.z * v.z + v.w * v.w);
  }

  const float wsum = wave_reduce_wmma(sum);
  const float wsq  = wave_reduce_wmma(sq);

  __shared__ float r0[TPB / 32], r1[TPB / 32];
  const int wid = threadIdx.x >> 5, lane = threadIdx.x & 31;
  if (lane == 0) { r0[wid] = wsum; r1[wid] = wsq; }
  __syncthreads();
  if (threadIdx.x == 0) {
    float a = 0.0f, b = 0.0f;
#pragma unroll
    for (int w = 0; w < TPB / 32; ++w) { a += r0[w]; b += r1[w]; }
    atomicAdd(&ws[WS_SUM + s], a);   // 64 atomics per address across grid
    atomicAdd(&ws[WS_SQ  + s], b);
  }
}

// Pass 2: mean / rstd per sample (one wave)
__global__ void k_finalize_stats(float* __restrict__ ws) {
  const int s = threadIdx.x;
  if (s < NB) {
    const float inv_n = 1.0f / (float)CHW;
    const float mean  = ws[WS_SUM + s] * inv_n;
    const float var   = ws[WS_SQ + s] * inv_n - mean * mean;
    ws[WS_MEAN + s] = mean;
    ws[WS_RSTD + s] = rsqrtf(var + EPS);
  }
}

// Pass 3: normalize teacher (L2-resident from pass 1), smooth-L1 vs student,
// reduce to per-sample loss accumulators.
__global__ void k_loss(const float4* __restrict__ t, const float4* __restrict__ x,
                       float* __restrict__ ws) {
  const int s = blockIdx.x / BPS;
  const int j = blockIdx.x % BPS;
  const float mean = ws[WS_MEAN + s];
  const float rstd = ws[WS_RSTD + s];
  const int base = s * V4_PER_SAMPLE + j * V4_PER_BLOCK + (int)threadIdx.x;

  float acc = 0.0f;
#pragma unroll
  for (int k = 0; k < ITERS; ++k) {
    const float4 tv = t[base + k * TPB];
    const float4 xv = x[base + k * TPB];
    float d;
    d = fabsf((tv.x - mean) * rstd - xv.x); acc += (d <= BETA) ? 0.25f * d * d : d - 1.0f;
    d = fabsf((tv.y - mean) * rstd - xv.y); acc += (d <= BETA) ? 0.25f * d * d : d - 1.0f;
    d = fabsf((tv.z - mean) * rstd - xv.z); acc += (d <= BETA) ? 0.25f * d * d : d - 1.0f;
    d = fabsf((tv.w - mean) * rstd - xv.w); acc += (d <= BETA) ? 0.25f * d * d : d - 1.0f;
  }

  const float wsum = wave_reduce_wmma(acc);

  __shared__ float r0[TPB / 32];
  const int wid = threadIdx.x >> 5, lane = threadIdx.x & 31;
  if (lane == 0) r0[wid] = wsum;
  __syncthreads();
  if (threadIdx.x == 0) {
    float a = 0.0f;
#pragma unroll
    for (int w = 0; w < TPB / 32; ++w) a += r0[w];
    atomicAdd(&ws[WS_LOSS + s], a);
  }
}

// Pass 4: mean over batch -> single scalar (one wave, WMMA wave reduction)
__global__ void k_write_out(const float* __restrict__ ws, float* __restrict__ out) {
  const float v = ws[WS_LOSS + (int)threadIdx.x];  // exactly 32 lanes
  const float total = wave_reduce_wmma(v);
  if (threadIdx.x == 0) out[0] = total * (1.0f / (float)NB);  // LOSS_WEIGHT = 1
}

// ---------------------------------------------------------------------------
extern "C" void kernel_launch(void* const* d_in, const int* in_sizes, int n_in,
                              void* d_out, int out_size, void* d_ws, size_t ws_size,
                              hipStream_t stream) {
  (void)in_sizes; (void)n_in; (void)out_size; (void)ws_size;
  const float4* teacher = (const float4*)d_in[0];
  const float4* student = (const float4*)d_in[1];
  float* ws  = (float*)d_ws;
  float* out = (float*)d_out;

  k_init_ws<<<1, 256, 0, stream>>>(ws);
  k_stats<<<NB * BPS, TPB, 0, stream>>>(teacher, ws);
  k_finalize_stats<<<1, 32, 0, stream>>>(ws);
  k_loss<<<NB * BPS, TPB, 0, stream>>>(teacher, student, ws);
  k_write_out<<<1, 32, 0, stream>>>(ws, out);
}